// MultiSMBlock_44753559224609
// MI455X (gfx1250) — compile-verified
//
#include <hip/hip_runtime.h>
#include <hip/hip_bf16.h>

// ---------------- problem constants ----------------
static constexpr int Hd  = 512;
static constexpr int Wd  = 512;
static constexpr int NCH = 8;     // channels
static constexpr int BSZ = 8;     // batch
// tile geometry: 32 wide x 4 tall = 128 pixels per block
static constexpr int TW  = 32;
static constexpr int THt = 4;
static constexpr int PIX = TW * THt;        // 128
static constexpr int XW  = TW + 2;          // 34 (halo)
static constexpr int XH  = THt + 2;         // 6
static constexpr int XSZ = XW * XH;         // 204

typedef __attribute__((ext_vector_type(2))) float v2f;
typedef __attribute__((ext_vector_type(8))) float v8f;

// generic pointer to LDS: low 32 bits == wave-relative LDS byte offset
__device__ __forceinline__ unsigned lds_off(const void* p) {
    return (unsigned)(unsigned long long)p;
}

// GVS-mode async copy: copies 4 bytes from global address saddr plus vaddr to the LDS address in vdst  (tracked by ASYNCcnt)
__device__ __forceinline__ void async_ld_b32(unsigned ldsOff, unsigned gOff,
                                             const float* base) {
    asm volatile("global_load_async_to_lds_b32 %0, %1, %2"
                 :: "v"(ldsOff), "v"(gOff), "s"(base)
                 : "memory");
}

__device__ __forceinline__ void wait_async_le2() {
#if __has_builtin(__builtin_amdgcn_s_wait_asynccnt)
    __builtin_amdgcn_s_wait_asynccnt(2);
#else
    asm volatile("s_wait_asynccnt 0x2" ::: "memory");
#endif
}
__device__ __forceinline__ void wait_async_le0() {
#if __has_builtin(__builtin_amdgcn_s_wait_asynccnt)
    __builtin_amdgcn_s_wait_asynccnt(0);
#else
    asm volatile("s_wait_asynccnt 0x0" ::: "memory");
#endif
}

// Issue one (b,c) x-halo-tile pair for iteration `it` into buffer `buf`.
// Every thread always issues exactly 2 async loads -> uniform ASYNCcnt/wave.
__device__ __forceinline__ void issue_xtile(int it, int buf, int t, int bx, int by,
                                            const float* __restrict__ x,
                                            float (*shx)[2][XSZ],
                                            float* shd) {
    const int c  = it >> 2;
    const int bb = it & 3;
#pragma unroll
    for (int r = 0; r < 2; ++r) {
        const int E  = t + 256 * r;           // 0..511 over [tile0 | tile1 | pad]
        int  ti      = E / XSZ;               // 0,1 valid; 2 = padding
        const bool valid = (ti < 2);
        const int tisel  = valid ? ti : 0;
        const int e   = E - ti * XSZ;
        const int esel = valid ? e : 0;
        const int b   = 2 * bb + tisel;       // batch index for this tile
        const int row = esel / XW;
        const int col = esel - row * XW;
        const int gy  = by * THt + row - 1;
        const int gx  = bx * TW  + col - 1;
        const bool inb = valid && ((unsigned)gy < (unsigned)Hd) &&
                                  ((unsigned)gx < (unsigned)Wd);
        const unsigned lo = inb ? lds_off(&shx[buf][tisel][esel])
                                : lds_off(&shd[t & 7]);
        const unsigned go = inb ? (unsigned)(((((b * NCH + c) * Hd) + gy) * Wd + gx) * 4)
                                : 0u;
        async_ld_b32(lo, go, x);
    }
}

__global__ __launch_bounds__(256)
void dynconv_fused_kernel(const float* __restrict__ image,
                          const float* __restrict__ x,
                          const float* __restrict__ conv_w,
                          const float* __restrict__ conv_b,
                          float* __restrict__ y) {
    __shared__ float sh_w[80 * 12];        // padded weight matrix (zeros outside 72x9)
    __shared__ float sh_b[80];             // padded bias
    __shared__ float sh_img[XSZ];          // image halo tile
    __shared__ float sh_K[72 * PIX];       // per-pixel kernels for this tile
    __shared__ float sh_x[2][2][XSZ];      // [dbl-buf][b-parity][halo tile]
    __shared__ float sh_dummy[8];          // sink for OOB async lanes

    const int t  = threadIdx.x;
    const int bx = blockIdx.x;
    const int by = blockIdx.y;

    // ---- stage weights / bias / image tile; pre-zero x buffers ----
    for (int i = t; i < 80 * 12; i += 256) {
        const int cm = i / 12, tp = i - cm * 12;
        sh_w[i] = (tp < 9 && cm < 72) ? conv_w[cm * 9 + tp] : 0.0f;
    }
    if (t < 80) sh_b[t] = (t < 72) ? conv_b[t] : 0.0f;
    if (t < XSZ) {
        const int row = t / XW, col = t - row * XW;
        const int gy = by * THt + row - 1, gx = bx * TW + col - 1;
        sh_img[t] = (((unsigned)gy < (unsigned)Hd) && ((unsigned)gx < (unsigned)Wd))
                        ? image[gy * Wd + gx] : 0.0f;
    }
    for (int i = t; i < 2 * 2 * XSZ; i += 256) ((float*)sh_x)[i] = 0.0f;
    if (t < 8) sh_dummy[t] = 0.0f;
    __syncthreads();

    // kick off the first x tile pair while we generate K
    issue_xtile(0, 0, t, bx, by, x, sh_x, sh_dummy);

    // ---- Phase 1: K[cm][p] = conv_w(72x9) @ im2col(image)(9xP) + bias ----
    // V_WMMA_F32_16X16X4_F32:  D(16x16) = A(16x4) * B(4x16) + C
    {
        const int wv   = t >> 5;      // wave -> N-tile (16 pixels)
        const int l    = t & 31;
        const int half = l >> 4;      // lane group: K-subchunk / M-offset selector
        const int n    = l & 15;
        const int p    = wv * 16 + n; // pixel 0..127
        const int ppx  = p & (TW - 1);
        const int ppy  = p >> 5;
#pragma unroll
        for (int mt = 0; mt < 5; ++mt) {          // cm tiles (80 rows padded)
            const int cmB = mt * 16;
            v8f c;
#pragma unroll
            for (int v = 0; v < 8; ++v) c[v] = sh_b[cmB + 8 * half + v];
#pragma unroll
            for (int k = 0; k < 3; ++k) {         // tap chunks (12 padded)
                const int t0 = 4 * k + 2 * half;  // lane-held taps t0, t0+1
                v2f a, bm;
                a.x = sh_w[(cmB + n) * 12 + t0];
                a.y = sh_w[(cmB + n) * 12 + t0 + 1];
                bm.x = (t0 < 9)
                     ? sh_img[(ppy + t0 / 3) * XW + ppx + (t0 % 3)] : 0.0f;
                bm.y = (t0 + 1 < 9)
                     ? sh_img[(ppy + (t0 + 1) / 3) * XW + ppx + ((t0 + 1) % 3)] : 0.0f;
                c = __builtin_amdgcn_wmma_f32_16x16x4_f32(
                        /*neg_a=*/false, a, /*neg_b=*/false, bm,
                        /*c_mod=*/(short)0, c, /*reuse_a=*/false, /*reuse_b=*/false);
            }
#pragma unroll
            for (int v = 0; v < 8; ++v) {         // C layout: M = v + 8*half, N = n
                const int cm = cmB + 8 * half + v;
                if (cm < 72) sh_K[cm * PIX + p] = c[v];
            }
        }
    }
    __syncthreads();

    // ---- Phase 2: y = sum_m K[c,m,p] * x[b,c, p+tap] with async double buffering ----
    const int hh  = t >> 7;           // 0/1: batch parity stream
    const int p   = t & 127;
    const int ppx = p & 31, ppy = p >> 5;
    const int gy  = by * THt + ppy, gx = bx * TW + ppx;

    float Kreg[9];
#pragma unroll 1
    for (int it = 0; it < 32; ++it) {
        const int cur = it & 1;
        const int c   = it >> 2;
        const int bb  = it & 3;
        __syncthreads();                          // buf[1-cur] free to overwrite
        if (it + 1 < 32) {
            issue_xtile(it + 1, 1 - cur, t, bx, by, x, sh_x, sh_dummy);
            wait_async_le2();                     // current tile's 2 loads landed
        } else {
            wait_async_le0();
        }
        __syncthreads();                          // all waves' loads visible
        if (bb == 0) {
#pragma unroll
            for (int m = 0; m < 9; ++m) Kreg[m] = sh_K[(c * 9 + m) * PIX + p];
        }
        const float* xt = sh_x[cur][hh];
        float acc = 0.0f;
#pragma unroll
        for (int i = 0; i < 3; ++i)
#pragma unroll
            for (int j = 0; j < 3; ++j)
                acc += Kreg[i * 3 + j] * xt[(ppy + i) * XW + ppx + j];
        const int b = 2 * bb + hh;
        y[((b * NCH + c) * Hd + gy) * Wd + gx] = acc;
    }
}

extern "C" void kernel_launch(void* const* d_in, const int* in_sizes, int n_in,
                              void* d_out, int out_size, void* d_ws, size_t ws_size,
                              hipStream_t stream) {
    (void)in_sizes; (void)n_in; (void)out_size; (void)d_ws; (void)ws_size;
    const float* image  = (const float*)d_in[0];
    const float* x      = (const float*)d_in[1];
    const float* conv_w = (const float*)d_in[2];
    const float* conv_b = (const float*)d_in[3];
    float* y = (float*)d_out;

    dim3 grid(Wd / TW, Hd / THt);   // 16 x 128 = 2048 blocks
    hipLaunchKernelGGL(dynconv_fused_kernel, grid, dim3(256), 0, stream,
                       image, x, conv_w, conv_b, y);
}